// GCNWithEdgeWeights_52218212385051
// MI455X (gfx1250) — compile-verified
//
#include <hip/hip_runtime.h>
#include <hip/hip_bf16.h>

typedef __attribute__((ext_vector_type(2))) float v2f;
typedef __attribute__((ext_vector_type(8))) float v8f;

#define NNODES 100000
#define NEDGES 1600000
#define DIN 64

// ---------------------------------------------------------------- zero
__global__ void zero_kernel(float* __restrict__ p, int n) {
    int i = blockIdx.x * blockDim.x + threadIdx.x;
    if (i < n) p[i] = 0.0f;
}

// ---------------------------------------------------------------- degrees
__global__ void degree_kernel(const int* __restrict__ src, const int* __restrict__ dst,
                              float* __restrict__ deg_out, float* __restrict__ deg_in, int E) {
    int e = blockIdx.x * blockDim.x + threadIdx.x;
    if (e >= E) return;
    unsafeAtomicAdd(&deg_out[src[e]], 1.0f);
    unsafeAtomicAdd(&deg_in[dst[e]], 1.0f);
}

// deg -> clip(deg,1)^-0.5  (in place, covers both arrays in one launch)
__global__ void norm_kernel(float* __restrict__ deg, int n) {
    int i = blockIdx.x * blockDim.x + threadIdx.x;
    if (i < n) deg[i] = rsqrtf(fmaxf(deg[i], 1.0f));
}

// ---------------------------------------------------------------- edge scatter
// agg[dst[e], :] += x[src[e], :] * norm_src[src[e]] * ew[e]   (D = 64)
// 16 threads per edge, 4 floats per thread.
__global__ void scatter_kernel(const float* __restrict__ x,
                               const int* __restrict__ src, const int* __restrict__ dst,
                               const float* __restrict__ ew, const float* __restrict__ ns,
                               float* __restrict__ agg, int E) {
    long gid = (long)blockIdx.x * blockDim.x + threadIdx.x;
    if (gid >= (long)E * 16) return;
    int e = (int)(gid >> 4);
    int f = (int)(gid & 15) << 2;           // feature offset 0..60
    int s = src[e];
    int d = dst[e];
    float coeff = ns[s] * ew[e];
    const float4 xv = *(const float4*)(x + (size_t)s * DIN + f);
    float* ap = agg + (size_t)d * DIN + f;
    unsafeAtomicAdd(ap + 0, xv.x * coeff);
    unsafeAtomicAdd(ap + 1, xv.y * coeff);
    unsafeAtomicAdd(ap + 2, xv.z * coeff);
    unsafeAtomicAdd(ap + 3, xv.w * coeff);
}

// ---------------------------------------------------------------- WMMA GEMM
// y[row, :N] = act( (agg[row, :64] * norm_dst[row]) @ W + b )
// One wave per 16x16 output tile; blockIdx.x = m-tile; wave id = n-tile.
// K = 64 -> 16 chained v_wmma_f32_16x16x4_f32.
template <int N, bool RELU>
__global__ void gemm_wmma_kernel(const float* __restrict__ agg,
                                 const float* __restrict__ norm_dst,
                                 const float* __restrict__ W,
                                 const float* __restrict__ bias,
                                 float* __restrict__ y) {
    const int lane = threadIdx.x & 31;
    const int ntile = threadIdx.x >> 5;       // which 16-col tile of N
    const int half = lane >> 4;               // 0: K pair {0,1}; 1: K pair {2,3}
    const int m = lane & 15;

    const int row = blockIdx.x * 16 + m;      // A row handled by this lane
    const int col = ntile * 16 + m;           // B/D column handled by this lane

    const float nd = norm_dst[row];
    const float* arow = agg + (size_t)row * 64;

    v8f c = {};
#pragma unroll
    for (int k0 = 0; k0 < 64; k0 += 4) {
        const int k = k0 + 2 * half;
        // A 16x4: lanes 0-15 hold K={k0,k0+1}, lanes 16-31 hold K={k0+2,k0+3}
        const float2 av = *(const float2*)(arow + k);
        v2f a;
        a.x = av.x * nd;
        a.y = av.y * nd;
        // B 4x16: lane = column, half selects K pair; W is [K][N] row-major
        v2f b;
        b.x = W[(size_t)k * N + col];
        b.y = W[(size_t)(k + 1) * N + col];
        c = __builtin_amdgcn_wmma_f32_16x16x4_f32(false, a, false, b,
                                                  (short)0, c, false, false);
    }

    const float bn = bias[col];
#pragma unroll
    for (int j = 0; j < 8; ++j) {
        // C/D: VGPR j -> row (j + 8*half), lane -> col
        const int r = blockIdx.x * 16 + j + 8 * half;
        float v = c[j] + bn;
        if (RELU) v = fmaxf(v, 0.0f);
        y[(size_t)r * N + col] = v;
    }
}

// ---------------------------------------------------------------- launch
extern "C" void kernel_launch(void* const* d_in, const int* in_sizes, int n_in,
                              void* d_out, int out_size, void* d_ws, size_t ws_size,
                              hipStream_t stream) {
    const float* features = (const float*)d_in[0];
    const float* ew       = (const float*)d_in[1];
    const int*   src      = (const int*)d_in[2];
    const int*   dst      = (const int*)d_in[3];
    const float* W1 = (const float*)d_in[4];
    const float* b1 = (const float*)d_in[5];
    const float* W2 = (const float*)d_in[6];
    const float* b2 = (const float*)d_in[7];
    const float* Wp = (const float*)d_in[8];
    const float* bp = (const float*)d_in[9];

    float* ws       = (float*)d_ws;
    float* norm_src = ws;                        // NNODES
    float* norm_dst = ws + NNODES;               // NNODES
    float* agg      = ws + 2 * NNODES;           // NNODES * 64
    float* xbuf     = agg + (size_t)NNODES * 64; // NNODES * 64

    const int E = NEDGES;
    const int feat_elems = NNODES * 64;
    const int mtiles = NNODES / 16;              // 6250, exact

    dim3 blk(256);
    auto blocks = [](long n) { return dim3((unsigned)((n + 255) / 256)); };

    // degrees -> norms
    zero_kernel<<<blocks(2 * NNODES), blk, 0, stream>>>(norm_src, 2 * NNODES);
    degree_kernel<<<blocks(E), blk, 0, stream>>>(src, dst, norm_src, norm_dst, E);
    norm_kernel<<<blocks(2 * NNODES), blk, 0, stream>>>(norm_src, 2 * NNODES);

    // ---- layer 1: features -> xbuf (relu), N = 64
    zero_kernel<<<blocks(feat_elems), blk, 0, stream>>>(agg, feat_elems);
    scatter_kernel<<<blocks((long)E * 16), blk, 0, stream>>>(features, src, dst, ew, norm_src, agg, E);
    gemm_wmma_kernel<64, true><<<mtiles, 128, 0, stream>>>(agg, norm_dst, W1, b1, xbuf);

    // ---- layer 2: xbuf -> xbuf (relu), N = 64
    zero_kernel<<<blocks(feat_elems), blk, 0, stream>>>(agg, feat_elems);
    scatter_kernel<<<blocks((long)E * 16), blk, 0, stream>>>(xbuf, src, dst, ew, norm_src, agg, E);
    gemm_wmma_kernel<64, true><<<mtiles, 128, 0, stream>>>(agg, norm_dst, W2, b2, xbuf);

    // ---- layer 3: xbuf -> d_out (no relu), N = 32
    zero_kernel<<<blocks(feat_elems), blk, 0, stream>>>(agg, feat_elems);
    scatter_kernel<<<blocks((long)E * 16), blk, 0, stream>>>(xbuf, src, dst, ew, norm_src, agg, E);
    gemm_wmma_kernel<32, false><<<mtiles, 64, 0, stream>>>(agg, norm_dst, Wp, bp, (float*)d_out);
}